// KNRM_25718264169296
// MI455X (gfx1250) — compile-verified
//
#include <hip/hip_runtime.h>
#include <hip/hip_bf16.h>

// ---------------- problem constants (from reference) ----------------
#define VOCAB   100000
#define EDIM    128
#define BATCH   512
#define QLEN    32
#define DLEN    256
#define KNUM    21
#define CHUNK   64            // doc rows per LDS chunk (4 chunks)
#define QSTRIDE 132           // padded LDS row stride (floats) -> conflict-free wmma reads
#define DSTRIDE 132
#define MMSTRIDE 65           // padded mm row stride (floats) -> conflict-free pooling reads

typedef __attribute__((ext_vector_type(2))) float v2f;
typedef __attribute__((ext_vector_type(8))) float v8f;

// mus = sorted([1.0, 0.95, 0.85, ..., -0.95])
__constant__ float c_MU[KNUM] = {
    -0.95f, -0.85f, -0.75f, -0.65f, -0.55f, -0.45f, -0.35f, -0.25f, -0.15f, -0.05f,
     0.05f,  0.15f,  0.25f,  0.35f,  0.45f,  0.55f,  0.65f,  0.75f,  0.85f,  0.95f, 1.0f };
// sigmas (with the reference's swapped call) = [0.1, 0.001 x20]; coef = -0.5/sigma^2
__constant__ float c_CF[KNUM] = {
    -50.0f,
    -500000.0f, -500000.0f, -500000.0f, -500000.0f, -500000.0f,
    -500000.0f, -500000.0f, -500000.0f, -500000.0f, -500000.0f,
    -500000.0f, -500000.0f, -500000.0f, -500000.0f, -500000.0f,
    -500000.0f, -500000.0f, -500000.0f, -500000.0f, -500000.0f };

__device__ __forceinline__ void load_norm_row(const float* __restrict__ emb,
                                              int idx, int lane,
                                              float* __restrict__ dst /* LDS row */)
{
    idx = min(max(idx, 0), VOCAB - 1);
    const float4 v = *(const float4*)(emb + (size_t)idx * EDIM + lane * 4);
    float ss = v.x * v.x + v.y * v.y + v.z * v.z + v.w * v.w;
    #pragma unroll
    for (int off = 16; off > 0; off >>= 1) ss += __shfl_xor(ss, off, 32);
    const float inv = 1.0f / fmaxf(sqrtf(ss), 1e-8f);
    float4 nv;
    nv.x = v.x * inv; nv.y = v.y * inv; nv.z = v.z * inv; nv.w = v.w * inv;
    *(float4*)(dst + lane * 4) = nv;
}

__global__ __launch_bounds__(256)
void knrm_pair_kernel(const float* __restrict__ emb,
                      const float* __restrict__ w1, const float* __restrict__ b1,
                      const float* __restrict__ w2, const float* __restrict__ b2,
                      const float* __restrict__ w3, const float* __restrict__ b3,
                      const int* __restrict__ q1i, const int* __restrict__ d1i,
                      const int* __restrict__ q2i, const int* __restrict__ d2i,
                      float* __restrict__ out)
{
    __shared__ float qs [QLEN  * QSTRIDE];   // normalized query rows
    __shared__ float dsh[CHUNK * DSTRIDE];   // normalized doc-chunk rows
    __shared__ float mmb[QLEN * MMSTRIDE];   // mm tile buffer [32][64] (padded)
    __shared__ float kout[KNUM];
    __shared__ float logits_sh[2];

    const int b    = blockIdx.x;
    const int tid  = threadIdx.x;
    const int lane = tid & 31;
    const int wave = tid >> 5;               // 8 waves

    const int qt = wave >> 2;                // 0..1 : 16-row query tile
    const int dt = wave & 3;                 // 0..3 : 16-col doc sub-tile in chunk
    const int rr = lane & 15;
    const int hh = lane >> 4;
    const int myq   = tid >> 3;              // 0..31
    const int dbase = (tid & 7) * 8;

    // A lane layout (f32 16x4): row = lane%16, VGPR0/1 = K {2h, 2h+1}.
    // B (f32 4x16) mirrors it with col = lane%16, so both read row-major LDS identically.
    const float* abase = &qs [(qt * 16 + rr) * QSTRIDE + 2 * hh];
    const float* bbase = &dsh[(dt * 16 + rr) * DSTRIDE + 2 * hh];

    for (int pred = 0; pred < 2; ++pred) {
        const int* qi = pred ? q2i : q1i;
        const int* di = pred ? d2i : d1i;

        // ---- 1) query rows -> LDS (each wave: 4 rows) ----
        for (int r = wave; r < QLEN; r += 8)
            load_norm_row(emb, qi[b * QLEN + r], lane, &qs[r * QSTRIDE]);
        __syncthreads();                     // qs ready

        // hoist the whole A strip (K = 0..127) into registers: invariant over chunks
        v2f areg[EDIM / 4];
        #pragma unroll
        for (int kk = 0; kk < EDIM / 4; ++kk)
            areg[kk] = *(const v2f*)(abase + 4 * kk);

        // per-thread pooling accumulators
        float acc[KNUM];
        #pragma unroll
        for (int k = 0; k < KNUM; ++k) acc[k] = 0.0f;

        // ---- 2) doc chunks ----
        for (int ch = 0; ch < 4; ++ch) {
            __syncthreads();                 // prev chunk: wmma dsh reads + pooling done
            #pragma unroll
            for (int i = 0; i < 8; ++i) {
                const int r = wave * 8 + i;  // row in chunk
                load_norm_row(emb, di[b * DLEN + ch * CHUNK + r], lane, &dsh[r * DSTRIDE]);
            }
            __syncthreads();                 // chunk resident

            // ---- 2a) WMMA, two independent accumulator chains over K ----
            v8f c0 = {}, c1 = {};
            #pragma unroll
            for (int kk = 0; kk < EDIM / 4; kk += 2) {
                v2f b0 = *(const v2f*)(bbase + 4 * kk);
                v2f b1v = *(const v2f*)(bbase + 4 * kk + 4);
                c0 = __builtin_amdgcn_wmma_f32_16x16x4_f32(
                        false, areg[kk],     false, b0,  (short)0, c0, false, false);
                c1 = __builtin_amdgcn_wmma_f32_16x16x4_f32(
                        false, areg[kk + 1], false, b1v, (short)0, c1, false, false);
            }
            // D layout: VGPR p -> M = p + 8*(lane>>4), N = lane%16
            #pragma unroll
            for (int p = 0; p < 8; ++p) {
                const int m = p + 8 * hh;
                mmb[(qt * 16 + m) * MMSTRIDE + dt * 16 + rr] = c0[p] + c1[p];
            }
            __syncthreads();                 // mm chunk ready

            // ---- 2b) Gaussian kernel pooling over this chunk's 64 d-columns ----
            #pragma unroll
            for (int j = 0; j < 8; ++j) {
                const float x = mmb[myq * MMSTRIDE + dbase + j];
                #pragma unroll
                for (int k = 0; k < KNUM; ++k) {
                    const float d = x - c_MU[k];
                    acc[k] += __expf(c_CF[k] * d * d);
                }
            }
        }

        // ---- 3) reduce the 8 lanes sharing a q-row (all within one wave) ----
        #pragma unroll
        for (int k = 0; k < KNUM; ++k) {
            float s = acc[k];
            s += __shfl_xor(s, 1, 32);
            s += __shfl_xor(s, 2, 32);
            s += __shfl_xor(s, 4, 32);
            acc[k] = s;                      // full sum over d for (myq, k)
        }

        if (tid < KNUM) kout[tid] = 0.0f;
        __syncthreads();
        if ((tid & 7) == 0) {                // one lane per q-row contributes log1p
            #pragma unroll
            for (int k = 0; k < KNUM; ++k)
                atomicAdd(&kout[k], log1pf(acc[k]));
        }
        __syncthreads();

        // ---- 4) tiny MLP (21 -> 10 -> 5 -> 1), relu before each matmul ----
        if (tid == 0) {
            float h1[10];
            #pragma unroll
            for (int j = 0; j < 10; ++j) {
                float s = b1[j];
                for (int k = 0; k < KNUM; ++k) s += fmaxf(kout[k], 0.0f) * w1[k * 10 + j];
                h1[j] = s;
            }
            float h2[5];
            #pragma unroll
            for (int j = 0; j < 5; ++j) {
                float s = b2[j];
                for (int k = 0; k < 10; ++k) s += fmaxf(h1[k], 0.0f) * w2[k * 5 + j];
                h2[j] = s;
            }
            float s = b3[0];
            for (int k = 0; k < 5; ++k) s += fmaxf(h2[k], 0.0f) * w3[k];
            logits_sh[pred] = s;
        }
        __syncthreads();
    }

    if (tid == 0) {
        const float dz = logits_sh[0] - logits_sh[1];
        out[b] = 1.0f / (1.0f + __expf(-dz));
    }
}

extern "C" void kernel_launch(void* const* d_in, const int* in_sizes, int n_in,
                              void* d_out, int out_size, void* d_ws, size_t ws_size,
                              hipStream_t stream)
{
    (void)in_sizes; (void)n_in; (void)d_ws; (void)ws_size; (void)out_size;
    const float* emb = (const float*)d_in[0];
    const float* w1  = (const float*)d_in[1];
    const float* b1  = (const float*)d_in[2];
    const float* w2  = (const float*)d_in[3];
    const float* b2  = (const float*)d_in[4];
    const float* w3  = (const float*)d_in[5];
    const float* b3  = (const float*)d_in[6];
    const int*   q1  = (const int*)d_in[7];
    const int*   d1  = (const int*)d_in[8];
    const int*   q2  = (const int*)d_in[9];
    const int*   d2  = (const int*)d_in[10];
    float* out = (float*)d_out;

    knrm_pair_kernel<<<BATCH, 256, 0, stream>>>(emb, w1, b1, w2, b2, w3, b3,
                                                q1, d1, q2, d2, out);
}